// EdgeAwareLoss_11811160064796
// MI455X (gfx1250) — compile-verified
//
#include <hip/hip_runtime.h>

// ---------------------------------------------------------------------------
// EdgeAwareLoss for MI455X (gfx1250, wave32).
// predictions: [B=16, C=19, H=512, W=512] f32
// targets:     [B, H, W] int (class ids 0..18)
// out:         scalar mean of CE * (1 + edge * (EDGE_WEIGHT-1))
//
// One block per image row. The 3-row target halo (3 x 514 ints) is staged
// into LDS with CDNA5 async global->LDS loads (ASYNCcnt), then the 3x3
// dilate/erode window and the target gather are served entirely from LDS.
// Predictions stream from HBM exactly once with NT temporal hints (no reuse,
// keep L2 for the target halo). Memory-bound: ~334 MB -> ~15 us floor.
// ---------------------------------------------------------------------------

namespace {
constexpr int kB = 16;
constexpr int kC = 19;
constexpr int kH = 512;
constexpr int kW = 512;
constexpr int kHW = kH * kW;          // 262144
constexpr int kNPix = kB * kHW;       // 4194304
constexpr int kCHW = kC * kHW;        // 4980736
constexpr float kEdgeWeight = 2.0f;
constexpr int kThreads = 256;
constexpr int kRows = kB * kH;        // 8192 blocks, one per row
constexpr int kTileC = kW + 2;        // 514 halo columns
}  // namespace

// CDNA5 async global->LDS load, GVS mode: mem = SADDR(64) + VADDR(32) ; LDS
// destination offset in a VGPR. Tracked with ASYNCcnt.
#define ASYNC_LOAD_B32_TO_LDS(ldsoff, voff, sbase)                          \
  asm volatile("global_load_async_to_lds_b32 %0, %1, %2"                    \
               :: "v"(ldsoff), "v"(voff), "s"(sbase) : "memory")

__global__ __launch_bounds__(kThreads)
void edge_loss_partial(const float* __restrict__ pred,
                       const int* __restrict__ tgt,
                       float* __restrict__ partial) {
  const int tid = threadIdx.x;
  const int row = blockIdx.x;          // 0 .. kRows-1
  const int b = row >> 9;              // row / kH
  const int h = row & (kH - 1);        // row % kH

  __shared__ int lab[3][kTileC];       // 3-row clamped target halo
  __shared__ float sm[kThreads];

  // Low 32 bits of a flat pointer into __shared__ == LDS byte offset
  // (ISA 10.2: the LDS address is the pointer's low 32 bits), which is what async VDST expects.
  const unsigned lds_base = (unsigned)(uintptr_t)(&lab[0][0]);

  // ---- stage target halo: rows h-1..h+1, cols -1..kW, clamped -------------
#pragma unroll
  for (int r = 0; r < 3; ++r) {
    const int gr = min(max(h - 1 + r, 0), kH - 1);
    const int rowbase = (b * kH + gr) * kW;
    for (int c = tid; c < kTileC; c += kThreads) {
      const int gc = min(max(c - 1, 0), kW - 1);
      const unsigned voff = (unsigned)(rowbase + gc) * 4u;
      const unsigned loff = lds_base + (unsigned)(r * kTileC + c) * 4u;
      ASYNC_LOAD_B32_TO_LDS(loff, voff, tgt);
    }
  }
  // Each wave waits for its own async loads, then the workgroup barrier makes
  // every wave's staged data visible to all.
  asm volatile("s_wait_asynccnt 0" ::: "memory");
  __syncthreads();

  float acc = 0.0f;

#pragma unroll
  for (int pi = 0; pi < 2; ++pi) {
    const int x = tid + pi * kThreads;   // 0 .. kW-1

    // ---- edge mask from LDS: max != min over the (clamped) 3x3 window -----
    int tmin = 0x7fffffff;
    int tmax = -0x7fffffff;
    int t = 0;
#pragma unroll
    for (int r = 0; r < 3; ++r) {
#pragma unroll
      for (int dc = 0; dc < 3; ++dc) {
        const int l = lab[r][x + dc];
        if (r == 1 && dc == 1) t = l;    // center target, register reuse
        tmin = min(tmin, l);
        tmax = max(tmax, l);
      }
    }
    const float wgt = (tmax != tmin) ? kEdgeWeight : 1.0f;

    // ---- 19-channel log-softmax, register resident ------------------------
    const int base = b * kCHW + h * kW + x;
    const float* __restrict__ pp = pred + base;

    float v[kC];
    float m = -3.402823466e+38f;
#pragma unroll
    for (int c = 0; c < kC; ++c) {
      // Streaming data, zero reuse: NT hint keeps L2 for the target halo.
      v[c] = __builtin_nontemporal_load(pp + c * kHW);
      m = fmaxf(m, v[c]);
    }

    // Target logit via cndmask chain (no dynamic register indexing).
    float xt = v[0];
#pragma unroll
    for (int c = 1; c < kC; ++c) xt = (t == c) ? v[c] : xt;

    float s = 0.0f;
#pragma unroll
    for (int c = 0; c < kC; ++c) s += __expf(v[c] - m);  // v_exp_f32

    const float ce = m + __logf(s) - xt;                 // v_log_f32
    acc += wgt * ce;
  }

  // ---- deterministic in-block tree reduction ------------------------------
  sm[tid] = acc;
  __syncthreads();
#pragma unroll
  for (int off = kThreads / 2; off > 0; off >>= 1) {
    if (tid < off) sm[tid] += sm[tid + off];
    __syncthreads();
  }
  if (tid == 0) partial[blockIdx.x] = sm[0];
}

__global__ __launch_bounds__(kThreads)
void edge_loss_reduce(const float* __restrict__ partial,
                      float* __restrict__ out,
                      int nblocks) {
  const int tid = threadIdx.x;
  float acc = 0.0f;
  // Fixed-order strided accumulation -> bitwise deterministic across replays.
  for (int i = tid; i < nblocks; i += kThreads) acc += partial[i];

  __shared__ float sm[kThreads];
  sm[tid] = acc;
  __syncthreads();
#pragma unroll
  for (int off = kThreads / 2; off > 0; off >>= 1) {
    if (tid < off) sm[tid] += sm[tid + off];
    __syncthreads();
  }
  if (tid == 0) out[0] = sm[0] * (1.0f / (float)kNPix);
}

extern "C" void kernel_launch(void* const* d_in, const int* in_sizes, int n_in,
                              void* d_out, int out_size, void* d_ws, size_t ws_size,
                              hipStream_t stream) {
  const float* pred = (const float*)d_in[0];
  const int*   tgt  = (const int*)d_in[1];
  float* out = (float*)d_out;
  float* partial = (float*)d_ws;  // kRows floats = 32 KB of scratch

  edge_loss_partial<<<kRows, kThreads, 0, stream>>>(pred, tgt, partial);
  edge_loss_reduce<<<1, kThreads, 0, stream>>>(partial, out, kRows);
}